// TensorizedQPC_88510686036627
// MI455X (gfx1250) — compile-verified
//
#include <hip/hip_runtime.h>
#include <cstdint>

// Sizes from the reference
#define F0v 4096
#define F1v 2048
#define F2v 512
#define H2v 4
#define F3v 256
#define KKv 32
#define CCv 10
#define BBv 64
#define NEGV (-1e30f)
#define PAD 34   // padded row stride (even: keeps b64 alignment; gcd(34,64)=2
                 // -> conflict-free strided access patterns)

typedef float v2f __attribute__((ext_vector_type(2)));
typedef float v8f __attribute__((ext_vector_type(8)));

// Sanity check on the DEVICE pass only: the host pass doesn't declare amdgcn
// builtins (calls inside __global__ bodies are handled by clang's deferred
// diagnostics, but #error is evaluated eagerly in both passes).
#if defined(__HIP_DEVICE_COMPILE__) && defined(__gfx1250__)
#if !__has_builtin(__builtin_amdgcn_wmma_f32_16x16x4_f32)
#error "expected gfx1250 f32 WMMA builtin (confirmed lowering in round 1)"
#endif
#endif

// ---------------------------------------------------------------------------
// Stage 1: Tucker layer.  One workgroup (256 thr = 8 waves) per f in [0,F1).
//   z[o,b]    = sum_{i,j} exp(W[i,j,o]) * ea[i,b] * eb[j,b]
//   x1[f,o,b] = log(z[o,b]) - log(S[o]) + ma[b] + mb[b],  S[o]=sum_ij exp(W)
// M=32,N=64,K=1024 GEMM via v_wmma_f32_16x16x4_f32; 8 waves <-> 2x4 tile grid.
// Weight row (fixed i, 32x32 = 4KB) exp'd into a double-buffered transposed
// LDS tile so each WMMA operand pair is one ds_load_b64.  exp(W) is applied
// without max-subtraction: logits ~ N(0,1) (f32 exp safe to x=88) and any max
// shift cancels between log(z) and log(S).
// ---------------------------------------------------------------------------
__global__ __launch_bounds__(256) void tucker_kernel(
    const float* __restrict__ leaf_ll,      // (F0, K, B)
    const float* __restrict__ w_logits,     // (F1, K, K, K)
    const int*   __restrict__ idx_tucker,   // (F1, 2)
    float*       __restrict__ x1)           // (F1, K, B)
{
  const int f   = blockIdx.x;
  const int tid = threadIdx.x;
  const int b   = tid & 63;    // b column handled in prologue
  const int q   = tid >> 6;    // k-quarter

  __shared__ __align__(16) float s_eat[BBv * PAD];      // exp(a-ma), [b][k]
  __shared__ __align__(16) float s_ebt[BBv * PAD];      // exp(b-mb), [b][k]
  __shared__ __align__(16) float s_wt[2][KKv * PAD];    // exp(W[i,:,:]), [o][j]
  __shared__ float s_S[KKv];                            // softmax denominators
  __shared__ float s_ma[BBv], s_mb[BBv];
  __shared__ float s_red[8 * BBv];

  const int ia = idx_tucker[2 * f + 0];
  const int ib = idx_tucker[2 * f + 1];
  const float* ap = leaf_ll + (size_t)ia * (KKv * BBv);
  const float* bp = leaf_ll + (size_t)ib * (KKv * BBv);
  const float* wbase = w_logits + (size_t)f * (KKv * KKv * KKv);

  // --- prologue: per-b max over k, exp into transposed LDS -----------------
  float va[8], vb[8];
  float mA = -3.4e38f, mB = -3.4e38f;
  for (int r = 0; r < 8; ++r) {
    int k = q * 8 + r;
    va[r] = ap[k * BBv + b];
    vb[r] = bp[k * BBv + b];
    mA = fmaxf(mA, va[r]);
    mB = fmaxf(mB, vb[r]);
  }
  s_red[q * BBv + b]       = mA;
  s_red[(4 + q) * BBv + b] = mB;
  if (tid < KKv) s_S[tid] = 0.0f;
  __syncthreads();
  if (q == 0) {
    float m0 = fmaxf(fmaxf(s_red[b], s_red[BBv + b]),
                     fmaxf(s_red[2 * BBv + b], s_red[3 * BBv + b]));
    float m1 = fmaxf(fmaxf(s_red[4 * BBv + b], s_red[5 * BBv + b]),
                     fmaxf(s_red[6 * BBv + b], s_red[7 * BBv + b]));
    s_ma[b] = m0;
    s_mb[b] = m1;
  }
  __syncthreads();
  {
    float m0 = s_ma[b], m1 = s_mb[b];
    for (int r = 0; r < 8; ++r) {
      int k = q * 8 + r;
      s_eat[b * PAD + k] = __expf(va[r] - m0);
      s_ebt[b * PAD + k] = __expf(vb[r] - m1);
    }
  }

  // --- weight-row fill mapping: one float4 per thread per row --------------
  // g = 4*tid .. 4*tid+3  ->  j = tid>>3 (same for all 4), o = (4*tid&31)+c
  const int jw = tid >> 3;
  const int o0 = (tid * 4) & 31;
  float regS[4] = {0.f, 0.f, 0.f, 0.f};   // per-thread S[o0+c] partials

  // fill row 0 into buffer 0
  {
    const float4 w4 = *(const float4*)(wbase + tid * 4);
    float e0 = __expf(w4.x), e1 = __expf(w4.y), e2 = __expf(w4.z), e3 = __expf(w4.w);
    s_wt[0][(o0 + 0) * PAD + jw] = e0;
    s_wt[0][(o0 + 1) * PAD + jw] = e1;
    s_wt[0][(o0 + 2) * PAD + jw] = e2;
    s_wt[0][(o0 + 3) * PAD + jw] = e3;
    regS[0] += e0; regS[1] += e1; regS[2] += e2; regS[3] += e3;
  }
  __syncthreads();

  // --- wave -> output tile mapping ----------------------------------------
  const int lane = tid & 31;
  const int wv   = tid >> 5;       // 0..7
  const int ot   = wv >> 2;        // o-tile (0..1)
  const int bt   = wv & 3;         // b-tile (0..3)
  const int hlf  = lane >> 4;      // lane half (selects K pair per ISA layout)
  const int nn   = lane & 15;      // A row (o) / B col (b) within tile
  const int oo   = ot * 16 + nn;   // absolute o for A operand rows
  const int ob   = bt * 16 + nn;   // absolute b for B operand cols

  v8f acc = {0.f, 0.f, 0.f, 0.f, 0.f, 0.f, 0.f, 0.f};

  const float* arow = &s_eat[ob * PAD];   // ea[:,ob] (transposed row)
  const float* brow = &s_ebt[ob * PAD];   // eb[:,ob]

  for (int i = 0; i < KKv; ++i) {
    const int p = i & 1;
    const float* wt = &s_wt[p][oo * PAD];
    const float eav = arow[i];

    // 8 k-steps of v_wmma_f32_16x16x4_f32; each operand pair = 1 ds_load_b64
    for (int j0 = 0; j0 < KKv; j0 += 4) {
      int k0 = j0 + 2 * hlf;       // ISA 16x4 layout: VGPR v holds K = 2*hlf+v
      v2f A = *(const v2f*)&wt[k0];
      v2f E = *(const v2f*)&brow[k0];
      v2f Bm;
      Bm[0] = E[0] * eav;
      Bm[1] = E[1] * eav;
      acc = __builtin_amdgcn_wmma_f32_16x16x4_f32(
          false, A, false, Bm, (short)0, acc, false, false);
    }

    // fill next row into the other buffer while this one computes
    if (i + 1 < KKv) {
      const float4 w4 = *(const float4*)(wbase + (i + 1) * (KKv * KKv) + tid * 4);
      float e0 = __expf(w4.x), e1 = __expf(w4.y), e2 = __expf(w4.z), e3 = __expf(w4.w);
      float* wn = &s_wt[1 - p][0];
      wn[(o0 + 0) * PAD + jw] = e0;
      wn[(o0 + 1) * PAD + jw] = e1;
      wn[(o0 + 2) * PAD + jw] = e2;
      wn[(o0 + 3) * PAD + jw] = e3;
      regS[0] += e0; regS[1] += e1; regS[2] += e2; regS[3] += e3;
      if (i + 2 < KKv && tid < 32) {   // prefetch the row after that
        __builtin_prefetch(wbase + (i + 2) * (KKv * KKv) + tid * 32, 0, 1);
      }
    }
    __syncthreads();
  }

  // fold per-thread S partials (once per block, ds_add_f32)
  atomicAdd(&s_S[o0 + 0], regS[0]);
  atomicAdd(&s_S[o0 + 1], regS[1]);
  atomicAdd(&s_S[o0 + 2], regS[2]);
  atomicAdd(&s_S[o0 + 3], regS[3]);
  __syncthreads();

  // --- epilogue: C/D layout lane(0-15)->N, VGPR r -> M=r (+8 for lane>=16) --
  const float maddb = s_ma[ob] + s_mb[ob];
  float* orow = x1 + (size_t)f * (KKv * BBv);
  for (int r = 0; r < 8; ++r) {
    int o = ot * 16 + r + 8 * hlf;
    orow[o * BBv + ob] = __logf(acc[r]) - __logf(s_S[o]) + maddb;
  }
}

// ---------------------------------------------------------------------------
// Stage 2: masked mixing.  One workgroup per f2.  Validity derived from
// idx < F1 (the padded zero row is referenced only by the masked slot).
// ---------------------------------------------------------------------------
__global__ __launch_bounds__(256) void mixing_kernel(
    const float* __restrict__ x1,          // (F1, K, B) in workspace
    const float* __restrict__ mix_logits,  // (F2, H2, K)
    const int*   __restrict__ idx_mix,     // (F2, H2)
    float*       __restrict__ x2)          // (F2, K, B)
{
  const int f   = blockIdx.x;
  const int tid = threadIdx.x;
  __shared__ int   s_idx[H2v];
  __shared__ float s_wm[H2v][KKv];

  if (tid < H2v) s_idx[tid] = idx_mix[f * H2v + tid];
  __syncthreads();

  if (tid < KKv) {  // per-k softmax over valid heads
    float l[H2v];
    float mx = -3.4e38f;
    for (int h = 0; h < H2v; ++h) {
      bool valid = s_idx[h] < F1v;
      l[h] = valid ? mix_logits[(f * H2v + h) * KKv + tid] : NEGV;
      mx = fmaxf(mx, l[h]);
    }
    float s = 0.f;
    for (int h = 0; h < H2v; ++h) {
      float e = (s_idx[h] < F1v) ? __expf(l[h] - mx) : 0.f;
      s_wm[h][tid] = e;
      s += e;
    }
    float inv = 1.0f / s;
    for (int h = 0; h < H2v; ++h) s_wm[h][tid] *= inv;
  }
  __syncthreads();

  const int b = tid & 63, q = tid >> 6;
  for (int r = 0; r < 8; ++r) {
    int k = q * 8 + r;
    float xv[H2v];
    float mmax = -3.4e38f;
    for (int h = 0; h < H2v; ++h) {
      if (s_idx[h] < F1v) {
        xv[h] = x1[((size_t)s_idx[h] * KKv + k) * BBv + b];
        mmax = fmaxf(mmax, xv[h]);
      }
    }
    float z = 0.f;
    for (int h = 0; h < H2v; ++h)
      if (s_idx[h] < F1v) z += __expf(xv[h] - mmax) * s_wm[h][k];
    x2[((size_t)f * KKv + k) * BBv + b] = __logf(z) + mmax;
  }
}

// ---------------------------------------------------------------------------
// Stage 3: CP head.  One workgroup per f3.  C=10 (< WMMA tile) and total
// work ~10M MACs, so plain VALU.
// ---------------------------------------------------------------------------
__global__ __launch_bounds__(256) void cp_kernel(
    const float* __restrict__ x2,         // (F2, K, B) in workspace
    const float* __restrict__ cp_logits,  // (F3, 2, K, C)
    const int*   __restrict__ idx_cp,     // (F3, 2)
    float*       __restrict__ out)        // (F3, C, B)
{
  const int f   = blockIdx.x;
  const int tid = threadIdx.x;
  __shared__ float s_e[2][KKv][BBv];   // exp(x - m)
  __shared__ float s_m[2][BBv];
  __shared__ float s_wc[2][KKv][CCv];  // softmax(cp_logits, axis=K)
  __shared__ float s_red[4 * BBv];
  __shared__ int   s_idx[2];

  if (tid < 2) s_idx[tid] = idx_cp[f * 2 + tid];
  __syncthreads();

  const int b = tid & 63, q = tid >> 6;
  for (int h = 0; h < 2; ++h) {
    const float* xp = x2 + (size_t)s_idx[h] * (KKv * BBv);
    float v[8];
    float m = -3.4e38f;
    for (int r = 0; r < 8; ++r) {
      v[r] = xp[(q * 8 + r) * BBv + b];
      m = fmaxf(m, v[r]);
    }
    s_red[q * BBv + b] = m;
    __syncthreads();
    if (q == 0)
      s_m[h][b] = fmaxf(fmaxf(s_red[b], s_red[BBv + b]),
                        fmaxf(s_red[2 * BBv + b], s_red[3 * BBv + b]));
    __syncthreads();
    float mm = s_m[h][b];
    for (int r = 0; r < 8; ++r) s_e[h][q * 8 + r][b] = __expf(v[r] - mm);
    __syncthreads();
  }

  if (tid < 2 * CCv) {  // softmax over K for each (h, o)
    int h = tid / CCv, o = tid % CCv;
    const float* lp = cp_logits + ((size_t)f * 2 + h) * (KKv * CCv) + o;
    float mx = -3.4e38f;
    for (int k = 0; k < KKv; ++k) mx = fmaxf(mx, lp[k * CCv]);
    float s = 0.f;
    for (int k = 0; k < KKv; ++k) {
      float e = __expf(lp[k * CCv] - mx);
      s_wc[h][k][o] = e;
      s += e;
    }
    float inv = 1.0f / s;
    for (int k = 0; k < KKv; ++k) s_wc[h][k][o] *= inv;
  }
  __syncthreads();

  for (int idx = tid; idx < CCv * BBv; idx += 256) {
    int o = idx >> 6, bb = idx & 63;
    float res = 0.f;
    for (int h = 0; h < 2; ++h) {
      float s = 0.f;
      for (int k = 0; k < KKv; ++k) s += s_e[h][k][bb] * s_wc[h][k][o];
      res += __logf(s) + s_m[h][bb];
    }
    out[((size_t)f * CCv + o) * BBv + bb] = res;
  }
}

// ---------------------------------------------------------------------------
extern "C" void kernel_launch(void* const* d_in, const int* in_sizes, int n_in,
                              void* d_out, int out_size, void* d_ws, size_t ws_size,
                              hipStream_t stream) {
  (void)in_sizes; (void)n_in; (void)out_size; (void)ws_size;
  const float* leaf_ll       = (const float*)d_in[0];
  const float* tucker_logits = (const float*)d_in[1];
  const float* mix_logits    = (const float*)d_in[2];
  const float* cp_logits     = (const float*)d_in[3];
  const int*   idx_tucker    = (const int*)d_in[4];
  const int*   idx_mix       = (const int*)d_in[5];
  // d_in[6] (mix_mask) intentionally unused: validity == (idx_mix < F1)
  const int*   idx_cp        = (const int*)d_in[7];

  float* x1 = (float*)d_ws;                       // 16 MB
  float* x2 = x1 + (size_t)F1v * KKv * BBv;       // +4 MB
  float* out = (float*)d_out;

  tucker_kernel<<<F1v, 256, 0, stream>>>(leaf_ll, tucker_logits, idx_tucker, x1);
  mixing_kernel<<<F2v, 256, 0, stream>>>(x1, mix_logits, idx_mix, x2);
  cp_kernel<<<F3v, 256, 0, stream>>>(x2, cp_logits, idx_cp, out);
}